// Model_88399016886763
// MI455X (gfx1250) — compile-verified
//
#include <hip/hip_runtime.h>

typedef __attribute__((ext_vector_type(2))) float v2f;
typedef __attribute__((ext_vector_type(8))) float v8f;

namespace {
constexpr int   kBS   = 1024;   // batch
constexpr int   kN    = 50;     // time points
constexpr int   kD    = 3;
constexpr int   kNV   = 150;    // 3*N
constexpr int   kNR   = 149;    // rows of A / size of G
constexpr int   kLD   = 152;    // padded row stride for A/G/B/Y/U
constexpr float kH    = 0.01f;  // STEP
}

// ---------------------------------------------------------------------------
// Stage one 128x32 A-slab and 32x64 B-slab into LDS.
// Fast path: GLOBAL_LOAD_ASYNC_TO_LDS_B32 (ASYNCcnt, no VGPR round trip).
// Edge path: guarded manual loads with zero fill.
// ---------------------------------------------------------------------------
__device__ __forceinline__ void stage_slab(float (*As)[33], float (*Bs)[65],
                                           const float* __restrict__ A, int lda,
                                           const float* __restrict__ B, int ldb,
                                           int M, int Nn, int K,
                                           int rowBase, int colBase, int k0, int tid)
{
    const bool fullTile = (k0 + 32 <= K) && (colBase + 64 <= Nn) && (rowBase + 128 <= M);
    if (fullTile) {
#pragma unroll
        for (int i = tid; i < 128 * 32; i += 256) {
            int r = i >> 5, c = i & 31;
            unsigned lds = (unsigned)(uintptr_t)&As[r][c];
            unsigned off = (unsigned)(((rowBase + r) * lda + k0 + c) * 4);
            asm volatile("global_load_async_to_lds_b32 %0, %1, %2"
                         :: "v"(lds), "v"(off), "s"(A) : "memory");
        }
#pragma unroll
        for (int i = tid; i < 32 * 64; i += 256) {
            int r = i >> 6, c = i & 63;
            unsigned lds = (unsigned)(uintptr_t)&Bs[r][c];
            unsigned off = (unsigned)(((k0 + r) * ldb + colBase + c) * 4);
            asm volatile("global_load_async_to_lds_b32 %0, %1, %2"
                         :: "v"(lds), "v"(off), "s"(B) : "memory");
        }
    } else {
        for (int i = tid; i < 128 * 32; i += 256) {
            int r = i >> 5, c = i & 31;
            int gr = rowBase + r, gk = k0 + c;
            As[r][c] = (gr < M && gk < K) ? A[(size_t)gr * lda + gk] : 0.f;
        }
        for (int i = tid; i < 32 * 64; i += 256) {
            int r = i >> 6, c = i & 63;
            int gk = k0 + r, gc = colBase + c;
            Bs[r][c] = (gk < K && gc < Nn) ? B[(size_t)gk * ldb + gc] : 0.f;
        }
    }
}

__device__ __forceinline__ void wait_async()
{
#if __has_builtin(__builtin_amdgcn_s_wait_asynccnt)
    __builtin_amdgcn_s_wait_asynccnt(0);
#else
    asm volatile("s_wait_asynccnt 0x0" ::: "memory");
#endif
}

// ---------------------------------------------------------------------------
// fp32 WMMA GEMM:  C(MxN) = act( A(MxK) @ B(KxN) + bias ),  row-major.
// 256 threads = 8 waves (4x2); each wave computes 32x32 (2x2 WMMA register
// blocking) -> 128x64 block tile.  K is software-pipelined in 32-slabs with
// double-buffered LDS: async-LDS DMA for slab k+1 overlaps WMMA on slab k.
// ---------------------------------------------------------------------------
__global__ __launch_bounds__(256)
void wmma_gemm_f32(const float* __restrict__ A, int lda,
                   const float* __restrict__ B, int ldb,
                   const float* __restrict__ bias,
                   float* __restrict__ C, int ldc,
                   int M, int Nn, int K, int relu)
{
    __shared__ float As[2][128][33];   // ~33.8 KB
    __shared__ float Bs[2][32][65];    // ~16.6 KB  (total ~50 KB of 320 KB/WGP)

    const int tid  = threadIdx.x;
    const int lane = tid & 31;
    const int wid  = tid >> 5;     // 0..7
    const int wy   = wid >> 1;     // 0..3 (32-row groups)
    const int wx   = wid & 1;      // 0..1 (32-col groups)
    const int half = lane >> 4;    // 0..1
    const int mn   = lane & 15;    // 0..15

    const int rowBase = blockIdx.y * 128;
    const int colBase = blockIdx.x * 64;

    v8f acc[2][2];
#pragma unroll
    for (int i = 0; i < 2; ++i)
#pragma unroll
        for (int j = 0; j < 2; ++j)
            acc[i][j] = (v8f){0.f, 0.f, 0.f, 0.f, 0.f, 0.f, 0.f, 0.f};

    // prologue: stage slab 0 into buffer 0
    stage_slab(As[0], Bs[0], A, lda, B, ldb, M, Nn, K, rowBase, colBase, 0, tid);
    wait_async();
    __syncthreads();

    int cur = 0;
    for (int k0 = 0; k0 < K; k0 += 32) {
        const int nxt = cur ^ 1;

        // issue async staging of the NEXT slab while we compute on the current
        if (k0 + 32 < K)
            stage_slab(As[nxt], Bs[nxt], A, lda, B, ldb, M, Nn, K,
                       rowBase, colBase, k0 + 32, tid);
        if (k0 + 64 < K) {  // deep L2 prefetch, two slabs ahead
            __builtin_prefetch(&A[(size_t)(rowBase + (tid >> 1)) * lda + k0 + 64], 0, 1);
            __builtin_prefetch(&B[(size_t)(k0 + 64 + (tid & 31)) * ldb + colBase], 0, 1);
        }

#pragma unroll
        for (int kk = 0; kk < 32; kk += 4) {
            const int kb = kk + 2 * half;   // ISA 16x4 f32 lane layout
            v2f a0, a1, b0, b1;
            a0.x = As[cur][wy * 32 + mn][kb];          a0.y = As[cur][wy * 32 + mn][kb + 1];
            a1.x = As[cur][wy * 32 + 16 + mn][kb];     a1.y = As[cur][wy * 32 + 16 + mn][kb + 1];
            b0.x = Bs[cur][kb][wx * 32 + mn];          b0.y = Bs[cur][kb + 1][wx * 32 + mn];
            b1.x = Bs[cur][kb][wx * 32 + 16 + mn];     b1.y = Bs[cur][kb + 1][wx * 32 + 16 + mn];
            acc[0][0] = __builtin_amdgcn_wmma_f32_16x16x4_f32(false, a0, false, b0, (short)0, acc[0][0], false, false);
            acc[0][1] = __builtin_amdgcn_wmma_f32_16x16x4_f32(false, a0, false, b1, (short)0, acc[0][1], false, false);
            acc[1][0] = __builtin_amdgcn_wmma_f32_16x16x4_f32(false, a1, false, b0, (short)0, acc[1][0], false, false);
            acc[1][1] = __builtin_amdgcn_wmma_f32_16x16x4_f32(false, a1, false, b1, (short)0, acc[1][1], false, false);
        }

        wait_async();       // my async writes into the next buffer are in LDS
        __syncthreads();    // everyone done reading cur + writing nxt
        cur = nxt;
    }

#pragma unroll
    for (int tj = 0; tj < 2; ++tj) {
        const int col = colBase + wx * 32 + tj * 16 + mn;
        if (col >= Nn) continue;
        float bv = bias ? bias[col] : 0.f;
#pragma unroll
        for (int ti = 0; ti < 2; ++ti) {
#pragma unroll
            for (int r = 0; r < 8; ++r) {
                int row = rowBase + wy * 32 + ti * 16 + r + 8 * half;  // ISA C/D layout
                if (row < M) {
                    float v = acc[ti][tj][r] + bv;
                    if (relu) v = fmaxf(v, 0.f);
                    C[(size_t)row * ldc + col] = v;
                }
            }
        }
    }
}

// ------------------------- small helper kernels ----------------------------

__global__ void dense_vec(const float* __restrict__ x, const float* __restrict__ W,
                          const float* __restrict__ b, float* __restrict__ y,
                          int Kin, int Nout, int relu)
{
    int o = blockIdx.x * blockDim.x + threadIdx.x;
    if (o >= Nout) return;
    float s = b[o];
    for (int k = 0; k < Kin; ++k) s += x[k] * W[(size_t)k * Nout + o];
    if (relu) s = fmaxf(s, 0.f);
    y[o] = s;
}

__global__ void finalize_xi(const float* __restrict__ xr, const float* __restrict__ mask,
                            float* __restrict__ ws_xi, float* __restrict__ out_xi)
{
    int i = threadIdx.x;
    if (i < 30) { float v = xr[i] * mask[i]; ws_xi[i] = v; out_xi[i] = v; }
}

__global__ void zero_buf(float* __restrict__ p, int n)
{
    int i = blockIdx.x * blockDim.x + threadIdx.x;
    if (i < n) p[i] = 0.f;
}

// A (149 x 150, stride kLD): constraint matrix (constant since STEP const)
__global__ void set_A(float* __restrict__ Ad)
{
    int t = blockIdx.x * blockDim.x + threadIdx.x;
    if (t < 50) {                                 // A_eq
        Ad[t * kLD + 3 * t + 1] = 1.f;
    } else if (t == 50) {                         // A_iv
        Ad[50 * kLD + 0] = 1.f;
    } else if (t < 100) {                         // A0
        int s = t - 51;
        Ad[t * kLD + 3 * (s + 1)] = 1.f;
        Ad[t * kLD + 3 * s]       = -1.f;
        Ad[t * kLD + 3 * s + 1]   = -kH;
        Ad[t * kLD + 3 * s + 2]   = -0.5f * kH * kH;
    } else if (t < 149) {                         // A1
        int s = t - 100;
        Ad[t * kLD + 3 * (s + 1) + 1] = 1.f;
        Ad[t * kLD + 3 * s + 1]       = -1.f;
        Ad[t * kLD + 3 * s + 2]       = -kH;
    }
}

// G = A A^T (149x149); also init GI = I
__global__ void build_G(const float* __restrict__ Ad, float* __restrict__ G,
                        float* __restrict__ GI)
{
    int idx = blockIdx.x * blockDim.x + threadIdx.x;
    if (idx >= kNR * kNR) return;
    int i = idx / kNR, j = idx % kNR;
    const float* ai = &Ad[i * kLD];
    const float* aj = &Ad[j * kLD];
    float s = 0.f;
    for (int k = 0; k < kNV; ++k) s += ai[k] * aj[k];
    G[i * kLD + j]  = s;
    GI[i * kLD + j] = (i == j) ? 1.f : 0.f;
}

// Single-block Gauss-Jordan: GI <- G^{-1}   (G is SPD; one-time ~3 MFLOP)
__global__ void gauss_jordan(float* __restrict__ G, float* __restrict__ GI)
{
    __shared__ float invp;
    for (int col = 0; col < kNR; ++col) {
        if (threadIdx.x == 0) invp = 1.f / G[col * kLD + col];
        __syncthreads();
        float ip = invp;
        for (int j = threadIdx.x; j < kNR; j += blockDim.x) {
            G[col * kLD + j]  *= ip;
            GI[col * kLD + j] *= ip;
        }
        __syncthreads();
        for (int r = threadIdx.x; r < kNR; r += blockDim.x) {
            if (r == col) continue;
            float f = G[r * kLD + col];
            if (f != 0.f) {
                for (int j = 0; j < kNR; ++j) {
                    G[r * kLD + j]  -= f * G[col * kLD + j];
                    GI[r * kLD + j] -= f * GI[col * kLD + j];
                }
            }
        }
        __syncthreads();
    }
}

// basis (poly library) + rhs einsum:  rhs[b][d][n] = sum_k basis[b,n,k]*xi[k,d]
__global__ void basis_rhs(const float* __restrict__ var, const float* __restrict__ xi,
                          float* __restrict__ rhs)
{
    int idx = blockIdx.x * blockDim.x + threadIdx.x;
    if (idx >= kBS * kN) return;
    int b = idx / kN, n = idx % kN;
    const float* v = &var[((size_t)b * kN + n) * kD];
    float v0 = v[0], v1 = v[1], v2 = v[2];
    float p[10] = {1.f, v0, v1, v2, v0 * v0, v0 * v1, v0 * v2, v1 * v1, v1 * v2, v2 * v2};
#pragma unroll
    for (int d = 0; d < kD; ++d) {
        float s = 0.f;
#pragma unroll
        for (int k = 0; k < 10; ++k) s += p[k] * xi[k * kD + d];
        rhs[((size_t)(b * kD + d)) * kN + n] = s;
    }
}

// Stack rhs/iv into B (3072 x kLD):  [rhs(50) | iv(1) | zeros]
__global__ void bmat_fill(const float* __restrict__ rhs, const float* __restrict__ var,
                          float* __restrict__ Bm)
{
    int idx = blockIdx.x * blockDim.x + threadIdx.x;
    if (idx >= kBS * kD * kLD) return;
    int row = idx / kLD, col = idx % kLD;
    float v = 0.f;
    if (col < kN) v = rhs[(size_t)row * kN + col];
    else if (col == kN) {
        int b = row / kD, d = row % kD;
        v = var[((size_t)b * kN + 0) * kD + d];      // iv = var[b,0,d]
    }
    Bm[idx] = v;
}

// eps[sys] = || A u - b ||_2  (block per system, 149 dense rows of length 150)
__global__ __launch_bounds__(256)
void eps_kernel(const float* __restrict__ Ad, const float* __restrict__ U,
                const float* __restrict__ Bm, float* __restrict__ eps)
{
    __shared__ float red[256];
    int sys = blockIdx.x;
    float acc = 0.f;
    for (int i = threadIdx.x; i < kNR; i += blockDim.x) {
        const float* a = &Ad[i * kLD];
        const float* u = &U[(size_t)sys * kLD];
        float s = -Bm[(size_t)sys * kLD + i];
        for (int k = 0; k < kNV; ++k) s += a[k] * u[k];
        acc += s * s;
    }
    red[threadIdx.x] = acc;
    __syncthreads();
    for (int off = 128; off > 0; off >>= 1) {
        if (threadIdx.x < off) red[threadIdx.x] += red[threadIdx.x + off];
        __syncthreads();
    }
    if (threadIdx.x == 0) eps[sys] = sqrtf(red[0]);
}

// out_x0[b, n, d] = U[(b*3+d)][3*n]   (x0.transpose(0,2,1))
__global__ void out_x0_kernel(const float* __restrict__ U, float* __restrict__ out)
{
    int idx = blockIdx.x * blockDim.x + threadIdx.x;
    if (idx >= kBS * kN * kD) return;
    int b = idx / (kN * kD);
    int rem = idx % (kN * kD);
    int n = rem / kD, d = rem % kD;
    out[idx] = U[((size_t)(b * kD + d)) * kLD + 3 * n];
}

__global__ void fill_steps(float* __restrict__ p, int n)
{
    int i = blockIdx.x * blockDim.x + threadIdx.x;
    if (i < n) p[i] = kH;
}

// ---------------------------------------------------------------------------
extern "C" void kernel_launch(void* const* d_in, const int* in_sizes, int n_in,
                              void* d_out, int out_size, void* d_ws, size_t ws_size,
                              hipStream_t stream)
{
    const float* net_iv   = (const float*)d_in[0];
    const float* param_in = (const float*)d_in[1];
    const float* pW1 = (const float*)d_in[2];
    const float* pb1 = (const float*)d_in[3];
    const float* pW2 = (const float*)d_in[4];
    const float* pb2 = (const float*)d_in[5];
    const float* pW3 = (const float*)d_in[6];
    const float* pb3 = (const float*)d_in[7];
    const float* nW1 = (const float*)d_in[8];
    const float* nb1 = (const float*)d_in[9];
    const float* nW2 = (const float*)d_in[10];
    const float* nb2 = (const float*)d_in[11];
    const float* nW3 = (const float*)d_in[12];
    const float* nb3 = (const float*)d_in[13];
    const float* mask = (const float*)d_in[14];

    float* out       = (float*)d_out;
    float* out_x0    = out;                 // 1024*50*3 = 153600
    float* out_steps = out + 153600;        // 1024*3*49 = 150528
    float* out_eps   = out + 304128;        // 1024*3    = 3072
    float* out_var   = out + 307200;        // 1024*50*3 = 153600
    float* out_xi    = out + 460800;        // 30

    float* ws     = (float*)d_ws;
    float* ws_h1p = ws;                     // 1024
    float* ws_h2p = ws + 1024;              // 1024
    float* ws_xir = ws + 2048;              // 32
    float* ws_xi  = ws + 2080;              // 32
    float* ws_A   = ws + 2112;              // 149*152 = 22648
    float* ws_G   = ws_A + 22648;
    float* ws_GI  = ws_G + 22648;
    float* ws_rhs = ws_GI + 22648;          // 3072*50 = 153600
    float* ws_B   = ws_rhs + 153600;        // 3072*152 = 466944
    float* ws_H1  = ws_B + 466944;          // 1024*1024 (later reused as Y)
    float* ws_H2  = ws_H1 + 1048576;        // 1024*1024 (later reused as U)

    // --- tiny param MLP: xi = mlp(param_in) * mask ---
    dense_vec<<<dim3(4), 256, 0, stream>>>(param_in, pW1, pb1, ws_h1p, 64, 1024, 1);
    dense_vec<<<dim3(4), 256, 0, stream>>>(ws_h1p, pW2, pb2, ws_h2p, 1024, 1024, 1);
    dense_vec<<<dim3(1), 32, 0, stream>>>(ws_h2p, pW3, pb3, ws_xir, 1024, 30, 0);
    finalize_xi<<<1, 32, 0, stream>>>(ws_xir, mask, ws_xi, out_xi);

    // --- big net MLP via fp32 WMMA GEMMs: var = mlp(net_iv) ---
    wmma_gemm_f32<<<dim3(16, 8), 256, 0, stream>>>(net_iv, 150, nW1, 1024, nb1,
                                                   ws_H1, 1024, 1024, 1024, 150, 1);
    wmma_gemm_f32<<<dim3(16, 8), 256, 0, stream>>>(ws_H1, 1024, nW2, 1024, nb2,
                                                   ws_H2, 1024, 1024, 1024, 1024, 1);
    wmma_gemm_f32<<<dim3(3, 8), 256, 0, stream>>>(ws_H2, 1024, nW3, 150, nb3,
                                                  out_var, 150, 1024, 150, 1024, 0);

    // --- constant ODE structure: A, G = A A^T, GI = G^{-1} (once per call) ---
    zero_buf<<<dim3(89), 256, 0, stream>>>(ws_A, kNR * kLD);
    set_A<<<1, 192, 0, stream>>>(ws_A);
    build_G<<<dim3(87), 256, 0, stream>>>(ws_A, ws_G, ws_GI);
    gauss_jordan<<<1, 256, 0, stream>>>(ws_G, ws_GI);

    // --- rhs / stacked b-vectors ---
    basis_rhs<<<dim3(200), 256, 0, stream>>>(out_var, ws_xi, ws_rhs);
    bmat_fill<<<dim3((kBS * kD * kLD) / 256), 256, 0, stream>>>(ws_rhs, out_var, ws_B);

    // --- batched solve as two WMMA GEMMs: Y = B @ G^{-1};  U = Y @ A ---
    wmma_gemm_f32<<<dim3(3, 24), 256, 0, stream>>>(ws_B, kLD, ws_GI, kLD, nullptr,
                                                   ws_H1, kLD, 3072, 149, 149, 0);
    wmma_gemm_f32<<<dim3(3, 24), 256, 0, stream>>>(ws_H1, kLD, ws_A, kLD, nullptr,
                                                   ws_H2, kLD, 3072, 150, 149, 0);

    // --- outputs ---
    eps_kernel<<<dim3(3072), 256, 0, stream>>>(ws_A, ws_H2, ws_B, out_eps);
    out_x0_kernel<<<dim3(600), 256, 0, stream>>>(ws_H2, out_x0);
    fill_steps<<<dim3(588), 256, 0, stream>>>(out_steps, 150528);
}